// FGNC_64965675319458
// MI455X (gfx1250) — compile-verified
//
#include <hip/hip_runtime.h>
#include <cstdint>
#include <cstddef>

// ---------------- problem constants (from reference) ----------------
#define HH    512
#define WW    512
#define BATCH 2
#define CHX   32     // x channels
#define GCH   16     // guide channels
#define EPSF  1e-6f
#define TH_L  0.2f
#define TH_HI 0.8f
#define S0F   4.5f

// ---------------- tiling ----------------
#define TS    16               // 16x16 output pixels per block
#define HT    (TS + 2)         // 18x18 halo tile
#define NPIX  (HT * HT)        // 324 halo pixels
#define SX_STRIDE 36           // padded x stride   (16B-aligned rows, conflict-free b128)
#define SG_STRIDE 20           // padded guide stride

// LDS layout (float offsets)
#define SX_OFF  0
#define SG_OFF  (NPIX * SX_STRIDE)            // 11664
#define SD_OFF  (SG_OFF + NPIX * SG_STRIDE)   // 18144  (raw d -> scaled log(d+eps))
#define SC_OFF  (SD_OFF + NPIX)               // 18468  (raw c, kept for C_new center)
#define SCS_OFF (SC_OFF + NPIX)               // 18792  (clip((c-.2)/.6)+eps)
#define LDS_FLOATS (SCS_OFF + NPIX)           // 19116 floats = 76464 B

// CDNA5 async global->LDS copy (per-lane gather, ASYNCcnt-tracked)
__device__ __forceinline__ void async_ld_b32(uint32_t lds_byte_addr, const float* gaddr) {
    asm volatile("global_load_async_to_lds_b32 %0, %1, off"
                 :: "v"(lds_byte_addr), "v"(gaddr)
                 : "memory");
}

extern "C" __global__ __launch_bounds__(256)
void fgnc_fused_kernel(const float* __restrict__ xin,
                       const float* __restrict__ dlight,
                       const float* __restrict__ cin,
                       const float* __restrict__ guide,
                       const float* __restrict__ lsf_p,
                       const float* __restrict__ lsd_p,
                       float* __restrict__ yout,
                       float* __restrict__ cout)
{
    extern __shared__ float smem[];
    float* sx  = smem + SX_OFF;
    float* sg  = smem + SG_OFF;
    float* sd  = smem + SD_OFF;
    float* sc  = smem + SC_OFF;
    float* scs = smem + SCS_OFF;
    // low 32 bits of the flat shared pointer == wave-relative LDS byte offset
    const uint32_t lds_base = (uint32_t)(uintptr_t)smem;

    const int b   = blockIdx.z;
    const int x0  = blockIdx.x * TS;
    const int y0  = blockIdx.y * TS;
    const int tid = threadIdx.x;

    const float* xb = xin    + (size_t)b * CHX * HH * WW;
    const float* gb = guide  + (size_t)b * GCH * HH * WW;
    const float* db = dlight + (size_t)b * HH * WW;
    const float* cb = cin    + (size_t)b * HH * WW;

    // scalars (uniform scalar loads; latency hidden behind the async fill)
    const float sigf   = __expf(lsf_p[0]) + EPSF;
    const float sigd   = __expf(lsd_p[0]) + EPSF;
    const float inv_sf = 2.0f / sigf;   // BETA = 2 folded into exponent
    const float inv_sd = 2.0f / sigd;   // GAMMA = 2 folded into exponent

    // ---------- stage haloed tile into LDS via async DMA ----------
    for (int idx = tid; idx < NPIX; idx += 256) {
        const int hy = idx / HT, hx = idx % HT;
        const int gy = y0 - 1 + hy, gx = x0 - 1 + hx;
        const bool inb = (gy >= 0) && (gy < HH) && (gx >= 0) && (gx < WW);
        const int gpix = gy * WW + gx;
        if (inb) {
            async_ld_b32(lds_base + (uint32_t)(SD_OFF + idx) * 4u, db + gpix);
            async_ld_b32(lds_base + (uint32_t)(SC_OFF + idx) * 4u, cb + gpix);
            #pragma unroll
            for (int ch = 0; ch < GCH; ++ch)
                async_ld_b32(lds_base + (uint32_t)(SG_OFF + idx * SG_STRIDE + ch) * 4u,
                             gb + (size_t)ch * HH * WW + gpix);
            #pragma unroll
            for (int ch = 0; ch < CHX; ++ch)
                async_ld_b32(lds_base + (uint32_t)(SX_OFF + idx * SX_STRIDE + ch) * 4u,
                             xb + (size_t)ch * HH * WW + gpix);
        } else {
            // zero-padding taps participate in the weights (jnp.pad semantics)
            sd[idx] = 0.0f;
            sc[idx] = 0.0f;
            #pragma unroll
            for (int ch = 0; ch < GCH; ++ch) sg[idx * SG_STRIDE + ch] = 0.0f;
            #pragma unroll
            for (int ch = 0; ch < CHX; ++ch) sx[idx * SX_STRIDE + ch] = 0.0f;
        }
    }
    asm volatile("s_wait_asynccnt 0x0" ::: "memory");
    __syncthreads();

    // ---------- per-halo-pixel transforms (shared by up to 9 consumers) ----------
    // sd  <- log(d+eps) * (2/sigma_d)   (pre-scaled so taps need only |a-b|)
    // scs <- clip((c-thL)/(thH-thL),0,1) + eps
    for (int idx = tid; idx < NPIX; idx += 256) {
        sd[idx]  = __logf(sd[idx] + EPSF) * inv_sd;
        scs[idx] = fminf(fmaxf((sc[idx] - TH_L) * (1.0f / (TH_HI - TH_L)), 0.0f), 1.0f) + EPSF;
    }
    __syncthreads();

    // ---------- per-pixel fused bilateral compute ----------
    const int tx = tid % TS, ty = tid / TS;
    const int pc = (ty + 1) * HT + (tx + 1);

    float gc[GCH];
    #pragma unroll
    for (int m = 0; m < GCH / 4; ++m) {
        float4 v = *reinterpret_cast<const float4*>(&sg[pc * SG_STRIDE + 4 * m]);
        gc[4 * m + 0] = v.x; gc[4 * m + 1] = v.y;
        gc[4 * m + 2] = v.z; gc[4 * m + 3] = v.w;
    }
    const float logd_c = sd[pc];   // already scaled by 2/sigma_d
    const float cc     = sc[pc];

    float A[9];
    float sumA = 0.0f;
    #pragma unroll
    for (int k = 0; k < 9; ++k) {
        const int ky = k / 3, kx = k % 3;
        const int pn = (ty + ky) * HT + (tx + kx);
        if (k == 4) {
            // center tap: g_diff == 0 and d_diff == 0 exactly -> A = C_sat + eps
            A[4] = scs[pc];
            sumA += A[4];
            continue;
        }
        float gdiff = 0.0f;
        #pragma unroll
        for (int m = 0; m < GCH / 4; ++m) {
            float4 v = *reinterpret_cast<const float4*>(&sg[pn * SG_STRIDE + 4 * m]);
            gdiff += fabsf(v.x - gc[4 * m + 0]) + fabsf(v.y - gc[4 * m + 1])
                   + fabsf(v.z - gc[4 * m + 2]) + fabsf(v.w - gc[4 * m + 3]);
        }
        const float ddiff = fabsf(sd[pn] - logd_c);              // pre-scaled
        // A_rgb^2 * A_depth^2 * A_flux  (ALPHA=1)
        const float a = __expf(-(gdiff * inv_sf + ddiff)) * scs[pn];
        A[k] = a;
        sumA += a;
    }
    const float inv_sum = 1.0f / (sumA + EPSF);

    float4 acc[CHX / 4];
    #pragma unroll
    for (int m = 0; m < CHX / 4; ++m) acc[m] = make_float4(0.f, 0.f, 0.f, 0.f);

    #pragma unroll
    for (int ky = 0; ky < 3; ++ky) {
        #pragma unroll
        for (int kx = 0; kx < 3; ++kx) {
            const int pn   = (ty + ky) * HT + (tx + kx);
            const float wk = A[ky * 3 + kx] * inv_sum;
            #pragma unroll
            for (int m = 0; m < CHX / 4; ++m) {
                float4 v = *reinterpret_cast<const float4*>(&sx[pn * SX_STRIDE + 4 * m]);
                acc[m].x = fmaf(wk, v.x, acc[m].x);
                acc[m].y = fmaf(wk, v.y, acc[m].y);
                acc[m].z = fmaf(wk, v.z, acc[m].z);
                acc[m].w = fmaf(wk, v.w, acc[m].w);
            }
        }
    }

    const int gy = y0 + ty, gx = x0 + tx;
    const size_t opix = (size_t)gy * WW + gx;
    float* yb = yout + (size_t)b * CHX * HH * WW;
    #pragma unroll
    for (int m = 0; m < CHX / 4; ++m) {
        yb[(size_t)(4 * m + 0) * HH * WW + opix] = acc[m].x;
        yb[(size_t)(4 * m + 1) * HH * WW + opix] = acc[m].y;
        yb[(size_t)(4 * m + 2) * HH * WW + opix] = acc[m].z;
        yb[(size_t)(4 * m + 3) * HH * WW + opix] = acc[m].w;
    }

    // support = tanh(sumA/S0) via exp (sumA >= 0, numerically safe)
    const float t = __expf(-2.0f * (sumA * (1.0f / S0F)));
    const float support = (1.0f - t) / (1.0f + t);
    cout[(size_t)b * HH * WW + opix] = cc + support * (1.0f - cc);
}

extern "C" void kernel_launch(void* const* d_in, const int* in_sizes, int n_in,
                              void* d_out, int out_size, void* d_ws, size_t ws_size,
                              hipStream_t stream) {
    (void)in_sizes; (void)n_in; (void)out_size; (void)d_ws; (void)ws_size;
    const float* x   = (const float*)d_in[0];
    const float* dl  = (const float*)d_in[1];
    const float* c   = (const float*)d_in[2];
    const float* g   = (const float*)d_in[3];
    const float* lsf = (const float*)d_in[4];
    const float* lsd = (const float*)d_in[5];

    float* yout = (float*)d_out;                         // (2,32,512,512)
    float* cout = yout + (size_t)BATCH * CHX * HH * WW;  // (2,1,512,512)

    dim3 grid(WW / TS, HH / TS, BATCH);
    dim3 block(256);
    const size_t ldsBytes = (size_t)LDS_FLOATS * sizeof(float);
    hipLaunchKernelGGL(fgnc_fused_kernel, grid, block, ldsBytes, stream,
                       x, dl, c, g, lsf, lsd, yout, cout);
}